// MEGNetBlock_20194936226691
// MI455X (gfx1250) — compile-verified
//
#include <hip/hip_runtime.h>
#include <hip/hip_bf16.h>
#include <math.h>

// MEGNet block sizes (fixed by the reference)
#define NN 50000      // nodes
#define NE 250000     // edges (directed pairs are 2*NE)
#define NB 16         // graphs in batch
#define D  128        // n2
#define D1 256        // n1
#define RT 5          // row tiles per edge-update block (16 rows = 8 base edges each)
#define RTD 5         // row tiles per dense-pre block (big launches)

typedef __attribute__((ext_vector_type(2))) float v2f;
typedef __attribute__((ext_vector_type(8))) float v8f;

// SoftPlus2(x) = softplus(x) - log(2), numerically stable
__device__ __forceinline__ float sp2f(float x) {
  return fmaxf(x, 0.0f) + log1pf(__expf(-fabsf(x))) - 0.69314718055994531f;
}

// C/D f32 16x16 layout: lane l, vgpr v -> element (m = v + 8*(l>>4), n = l&15).
__device__ __forceinline__ v8f bias_acc(const float* __restrict__ bias, int col0, int lane) {
  const float bv = bias[col0 + (lane & 15)];
  v8f c;
#pragma unroll
  for (int v = 0; v < 8; ++v) c[v] = bv;
  return c;
}

// 16x16 tile at col0 of A[16][K] (LDS) x B[K][ldb] (global), B streamed per step.
// (Used only by the small node/global update GEMMs.)
__device__ __forceinline__ v8f wmma_kloop(const float* __restrict__ A, int K,
                                          const float* __restrict__ B, int ldb,
                                          int col0, int lane, v8f acc) {
  const int m  = lane & 15;
  const int kb = (lane >> 4) << 1;
  const int n  = lane & 15;
#pragma unroll 4
  for (int k = 0; k < K; k += 4) {
    v2f a, b;
    a.x = A[m * K + k + kb];
    a.y = A[m * K + k + kb + 1];
    b.x = B[(long long)(k + kb) * ldb + col0 + n];
    b.y = B[(long long)(k + kb + 1) * ldb + col0 + n];
    acc = __builtin_amdgcn_wmma_f32_16x16x4_f32(false, a, false, b, (short)0, acc,
                                                false, false);
  }
  return acc;
}

// Register-blocked multi-row-tile GEMM core: B-fragments for a K=64 chunk are
// preloaded into 32 VGPRs and swept over NT row tiles of A (LDS, row stride K).
template <int NT>
__device__ __forceinline__ void wmma_rb(const float* __restrict__ A, int K,
                                        const float* __restrict__ B, int ldb,
                                        int col0, int lane, v8f (&acc)[NT]) {
  const int m  = lane & 15;
  const int kb = (lane >> 4) << 1;
  const int n  = lane & 15;
  for (int kc = 0; kc < K; kc += 64) {
    v2f breg[16];
#pragma unroll
    for (int s = 0; s < 16; ++s) {
      const int k = kc + s * 4 + kb;
      breg[s].x = B[(long long)k * ldb + col0 + n];
      breg[s].y = B[(long long)(k + 1) * ldb + col0 + n];
    }
#pragma unroll
    for (int rt = 0; rt < NT; ++rt) {
      const float* Ar = &A[(rt * 16 + m) * K];
#pragma unroll
      for (int s = 0; s < 16; ++s) {
        const int k = kc + s * 4 + kb;
        v2f a;
        a.x = Ar[k];
        a.y = Ar[k + 1];
        acc[rt] = __builtin_amdgcn_wmma_f32_16x16x4_f32(
            false, a, false, breg[s], (short)0, acc[rt], false, false);
      }
    }
  }
}

// ---------------------------------------------------------------------------
// Zero the scatter accumulators.
__global__ void __launch_bounds__(256)
init_kernel(float* __restrict__ efm_sum, float* __restrict__ efm_cnt,
            float* __restrict__ em_sum, float* __restrict__ em_cnt,
            float* __restrict__ nm_sum, float* __restrict__ nm_cnt) {
  const long long i = (long long)blockIdx.x * 256 + threadIdx.x;
  if (i < (long long)NN * D) efm_sum[i] = 0.0f;
  if (i < NN) efm_cnt[i] = 0.0f;
  if (i < NB * D) { em_sum[i] = 0.0f; nm_sum[i] = 0.0f; }
  if (i < NB) { em_cnt[i] = 0.0f; nm_cnt[i] = 0.0f; }
}

// ---------------------------------------------------------------------------
// y = sp2(sp2(x @ W1 + b1) @ W2 + b2), x:[M,128], W1:[128,256], W2:[256,128].
// Block = NT row tiles of 16 rows; B chunks in registers, swept over tiles.
template <int NT>
__global__ void __launch_bounds__(128)
dense_pre_kernel(const float* __restrict__ x,
                 const float* __restrict__ W1, const float* __restrict__ b1,
                 const float* __restrict__ W2, const float* __restrict__ b2,
                 float* __restrict__ y) {
  __shared__ float A[NT * 16 * D];    // NT=5: 40 KB
  __shared__ float H[NT * 16 * D1];   // NT=5: 80 KB
  const int t = threadIdx.x;
  const int lane = t & 31;
  const int wave = t >> 5;
  const long long row0 = (long long)blockIdx.x * (NT * 16);
  const int n = lane & 15;

  for (int i = t; i < NT * 16 * D / 4; i += 128)
    ((float4*)A)[i] = ((const float4*)(x + row0 * D))[i];
  __syncthreads();

  // stage 1: [NT*16,128] @ [128,256] -> H, sp2
  for (int ct = wave; ct < D1 / 16; ct += 4) {
    const int col0 = ct * 16;
    v8f acc[NT];
#pragma unroll
    for (int rt = 0; rt < NT; ++rt) acc[rt] = bias_acc(b1, col0, lane);
    wmma_rb<NT>(A, D, W1, D1, col0, lane, acc);
#pragma unroll
    for (int rt = 0; rt < NT; ++rt)
#pragma unroll
      for (int v = 0; v < 8; ++v) {
        const int m = rt * 16 + v + ((lane >> 4) << 3);
        H[m * D1 + col0 + n] = sp2f(acc[rt][v]);
      }
  }
  __syncthreads();

  // stage 2: [NT*16,256] @ [256,128] -> y, sp2
  for (int ct = wave; ct < D / 16; ct += 4) {
    const int col0 = ct * 16;
    v8f acc[NT];
#pragma unroll
    for (int rt = 0; rt < NT; ++rt) acc[rt] = bias_acc(b2, col0, lane);
    wmma_rb<NT>(H, D1, W2, D, col0, lane, acc);
#pragma unroll
    for (int rt = 0; rt < NT; ++rt)
#pragma unroll
      for (int v = 0; v < 8; ++v) {
        const int m = rt * 16 + v + ((lane >> 4) << 3);
        y[(row0 + m) * D + col0 + n] = sp2f(acc[rt][v]);
      }
  }
}

// ---------------------------------------------------------------------------
// Edge update. Block = RT row-tiles of 16 rows; each tile holds 8 base edges:
// rows 0-7 forward, rows 8-15 reversed. The (e, E+e) fold then needs only a
// __shfl_xor(16) inside the wave (rows m and m+8 sit in lanes l and l+16),
// so out_ef is a plain store with the ef0 residual fused in.
__global__ void __launch_bounds__(128)
edge_update_kernel(const float* __restrict__ nf_pre, const float* __restrict__ ef_pre,
                   const float* __restrict__ gf_pre,
                   const int* __restrict__ edge_index, const int* __restrict__ batch,
                   const float* __restrict__ emW, const float* __restrict__ emb,
                   const float* __restrict__ ef0, float* __restrict__ out_ef,
                   float* __restrict__ efm_sum, float* __restrict__ efm_cnt,
                   float* __restrict__ em_sum, float* __restrict__ em_cnt) {
  __shared__ float A[RT * 16 * 4 * D];                    // 80 x 512 floats = 160 KB
  __shared__ int ssrc[RT * 16], sdst[RT * 16], sebm[RT * 16], smod[RT * 16];
  const int t = threadIdx.x;
  const int lane = t & 31;
  const int wave = t >> 5;
  const long long base0 = (long long)blockIdx.x * (RT * 8);

  // per-row metadata + mean denominators
  for (int r = t; r < RT * 16; r += 128) {
    const int rr = r & 15;
    const long long eb = base0 + (r >> 4) * 8 + (rr & 7);
    const int s0 = edge_index[eb];
    const int d0 = edge_index[NE + eb];
    const int s = (rr < 8) ? s0 : d0;
    const int d = (rr < 8) ? d0 : s0;
    const int g = batch[s];
    ssrc[r] = s; sdst[r] = d; sebm[r] = g; smod[r] = (int)eb;
    atomicAdd(&efm_cnt[d], 1.0f);
    atomicAdd(&em_cnt[g], 1.0f);
  }
  __syncthreads();

  // gather concat(nf[s], nf[d], ef, gf[g]) rows, float4 per thread per row
  for (int r = 0; r < RT * 16; ++r) {
    const int seg = t >> 5;
    const int off = (t & 31) * 4;
    const float* p;
    if (seg == 0)      p = nf_pre + (long long)ssrc[r] * D + off;
    else if (seg == 1) p = nf_pre + (long long)sdst[r] * D + off;
    else if (seg == 2) p = ef_pre + (long long)smod[r] * D + off;
    else               p = gf_pre + (long long)sebm[r] * D + off;
    *(float4*)&A[r * (4 * D) + t * 4] = *(const float4*)p;
  }
  __syncthreads();

  const int n = lane & 15;

  for (int ct = wave; ct < D / 16; ct += 4) {
    const int col0 = ct * 16;
    v8f acc[RT];
#pragma unroll
    for (int rt = 0; rt < RT; ++rt) acc[rt] = bias_acc(emb, col0, lane);
    wmma_rb<RT>(A, 4 * D, emW, D, col0, lane, acc);

#pragma unroll
    for (int rt = 0; rt < RT; ++rt) {
#pragma unroll
      for (int v = 0; v < 8; ++v) {
        const int mm = v + ((lane >> 4) << 3);
        const int r = rt * 16 + mm;
        const float val = acc[rt][v];
        const int col = col0 + n;
        atomicAdd(&efm_sum[(long long)sdst[r] * D + col], val);  // node mean num.
        atomicAdd(&em_sum[sebm[r] * D + col], val);              // graph mean num.
        const float other = __shfl_xor(val, 16, 32);             // partner row m^8
        if (lane < 16) {                                         // fold + residual
          const long long eb = smod[rt * 16 + v];
          out_ef[eb * D + col] = 0.5f * (val + other) + ef0[eb * D + col];
        }
      }
    }
  }
}

// ---------------------------------------------------------------------------
// Node update: concat(nf, mean-in-edges, gf[batch]) [16x384] @ nmW[384x128].
__global__ void __launch_bounds__(128)
node_update_kernel(const float* __restrict__ nf_pre,
                   const float* __restrict__ efm_sum, const float* __restrict__ efm_cnt,
                   const float* __restrict__ gf_pre, const int* __restrict__ batch,
                   const float* __restrict__ nmW, const float* __restrict__ nmb,
                   const float* __restrict__ nf0, float* __restrict__ out_nf,
                   float* __restrict__ nm_sum, float* __restrict__ nm_cnt) {
  __shared__ float A[16 * 3 * D];  // 24 KB
  __shared__ int sb[16];
  __shared__ float sinv[16];
  const int t = threadIdx.x;
  const int lane = t & 31;
  const int wave = t >> 5;
  const long long row0 = (long long)blockIdx.x * 16;

  if (t < 16) {
    const int g = batch[row0 + t];
    sb[t] = g;
    sinv[t] = 1.0f / fmaxf(efm_cnt[row0 + t], 1.0f);
    atomicAdd(&nm_cnt[g], 1.0f);
  }
  __syncthreads();

  for (int r = 0; r < 16; ++r) {
    if (t < 96) {
      const long long i = row0 + r;
      const int seg = t >> 5;
      const int off = (t & 31) * 4;
      float4 v;
      if (seg == 0)      v = *(const float4*)(nf_pre + i * D + off);
      else if (seg == 1) {
        v = *(const float4*)(efm_sum + i * D + off);
        const float inv = sinv[r];
        v.x *= inv; v.y *= inv; v.z *= inv; v.w *= inv;
      } else             v = *(const float4*)(gf_pre + (long long)sb[r] * D + off);
      *(float4*)&A[r * 3 * D + t * 4] = v;
    }
  }
  __syncthreads();

  for (int ct = wave; ct < D / 16; ct += 4) {
    const int col0 = ct * 16;
    v8f acc = bias_acc(nmb, col0, lane);
    acc = wmma_kloop(A, 3 * D, nmW, D, col0, lane, acc);
    const int n = lane & 15;
#pragma unroll
    for (int v = 0; v < 8; ++v) {
      const int m = v + ((lane >> 4) << 3);
      const float val = acc[v];
      const int col = col0 + n;
      const long long row = row0 + m;
      out_nf[row * D + col] = val + nf0[row * D + col];   // residual
      atomicAdd(&nm_sum[sb[m] * D + col], val);           // graph mean numerator
    }
  }
}

// ---------------------------------------------------------------------------
// Global update: B=16 is exactly one WMMA M-tile. One block.
__global__ void __launch_bounds__(128)
global_update_kernel(const float* __restrict__ em_sum, const float* __restrict__ em_cnt,
                     const float* __restrict__ nm_sum, const float* __restrict__ nm_cnt,
                     const float* __restrict__ gf_pre,
                     const float* __restrict__ gmW, const float* __restrict__ gmb,
                     const float* __restrict__ gf0, float* __restrict__ out_gf) {
  __shared__ float A[16 * 3 * D];
  const int t = threadIdx.x;
  const int lane = t & 31;
  const int wave = t >> 5;

  for (int r = 0; r < 16; ++r) {
    if (t < 96) {
      const int seg = t >> 5;
      const int off = (t & 31) * 4;
      float4 v;
      if (seg == 0) {
        v = *(const float4*)(em_sum + r * D + off);
        const float inv = 1.0f / fmaxf(em_cnt[r], 1.0f);
        v.x *= inv; v.y *= inv; v.z *= inv; v.w *= inv;
      } else if (seg == 1) {
        v = *(const float4*)(nm_sum + r * D + off);
        const float inv = 1.0f / fmaxf(nm_cnt[r], 1.0f);
        v.x *= inv; v.y *= inv; v.z *= inv; v.w *= inv;
      } else v = *(const float4*)(gf_pre + r * D + off);
      *(float4*)&A[r * 3 * D + t * 4] = v;
    }
  }
  __syncthreads();

  for (int ct = wave; ct < D / 16; ct += 4) {
    const int col0 = ct * 16;
    v8f acc = bias_acc(gmb, col0, lane);
    acc = wmma_kloop(A, 3 * D, gmW, D, col0, lane, acc);
    const int n = lane & 15;
#pragma unroll
    for (int v = 0; v < 8; ++v) {
      const int m = v + ((lane >> 4) << 3);
      const int col = col0 + n;
      out_gf[m * D + col] = acc[v] + gf0[m * D + col];
    }
  }
}

// ---------------------------------------------------------------------------
extern "C" void kernel_launch(void* const* d_in, const int* in_sizes, int n_in,
                              void* d_out, int out_size, void* d_ws, size_t ws_size,
                              hipStream_t stream) {
  (void)in_sizes; (void)n_in; (void)out_size; (void)ws_size;

  const float* node_features   = (const float*)d_in[0];
  const int*   edge_index      = (const int*)d_in[1];
  const float* edge_features   = (const float*)d_in[2];
  const float* global_features = (const float*)d_in[3];
  const int*   batch           = (const int*)d_in[4];
  const float* eW1 = (const float*)d_in[5],  *eb1 = (const float*)d_in[6];
  const float* eW2 = (const float*)d_in[7],  *eb2 = (const float*)d_in[8];
  const float* nW1 = (const float*)d_in[9],  *nb1 = (const float*)d_in[10];
  const float* nW2 = (const float*)d_in[11], *nb2 = (const float*)d_in[12];
  const float* gW1 = (const float*)d_in[13], *gb1 = (const float*)d_in[14];
  const float* gW2 = (const float*)d_in[15], *gb2 = (const float*)d_in[16];
  const float* emW = (const float*)d_in[17], *emb = (const float*)d_in[18];
  const float* nmW = (const float*)d_in[19], *nmb = (const float*)d_in[20];
  const float* gmW = (const float*)d_in[21], *gmb = (const float*)d_in[22];

  // workspace layout (floats)
  float* ws      = (float*)d_ws;
  float* nf_pre  = ws;                                   // N*D
  float* ef_pre  = nf_pre  + (size_t)NN * D;             // E*D
  float* gf_pre  = ef_pre  + (size_t)NE * D;             // B*D
  float* efm_sum = gf_pre  + (size_t)NB * D;             // N*D
  float* efm_cnt = efm_sum + (size_t)NN * D;             // N
  float* em_sum  = efm_cnt + NN;                         // B*D
  float* em_cnt  = em_sum  + NB * D;                     // B
  float* nm_sum  = em_cnt  + NB;                         // B*D
  float* nm_cnt  = nm_sum  + NB * D;                     // B

  // output layout: nf [N*D], ef [E*D], gf [B*D]
  float* out_nf = (float*)d_out;
  float* out_ef = out_nf + (size_t)NN * D;
  float* out_gf = out_ef + (size_t)NE * D;

  const long long initN = (long long)NN * D;
  init_kernel<<<(unsigned)((initN + 255) / 256), 256, 0, stream>>>(
      efm_sum, efm_cnt, em_sum, em_cnt, nm_sum, nm_cnt);

  dense_pre_kernel<RTD><<<NE / (RTD * 16), 128, 0, stream>>>(
      edge_features, eW1, eb1, eW2, eb2, ef_pre);
  dense_pre_kernel<RTD><<<NN / (RTD * 16), 128, 0, stream>>>(
      node_features, nW1, nb1, nW2, nb2, nf_pre);
  dense_pre_kernel<1><<<1, 128, 0, stream>>>(
      global_features, gW1, gb1, gW2, gb2, gf_pre);

  edge_update_kernel<<<NE / (RT * 8), 128, 0, stream>>>(
      nf_pre, ef_pre, gf_pre, edge_index, batch, emW, emb,
      edge_features, out_ef, efm_sum, efm_cnt, em_sum, em_cnt);

  node_update_kernel<<<NN / 16, 128, 0, stream>>>(
      nf_pre, efm_sum, efm_cnt, gf_pre, batch, nmW, nmb,
      node_features, out_nf, nm_sum, nm_cnt);

  global_update_kernel<<<1, 128, 0, stream>>>(
      em_sum, em_cnt, nm_sum, nm_cnt, gf_pre, gmW, gmb, global_features, out_gf);
}